// GMTPooling_21801253995181
// MI455X (gfx1250) — compile-verified
//
#include <hip/hip_runtime.h>
#include <hip/hip_bf16.h>
#include <math.h>

typedef __attribute__((ext_vector_type(2))) float v2f;
typedef __attribute__((ext_vector_type(8))) float v8f;

// Problem constants (fixed by the reference module)
#define EDIM 256
#define HNUM 4
#define BGRAPH 512
#define KCOMB 1024   // H * CIN

// ---------------------------------------------------------------------------
// Generic f32 WMMA GEMM: C[M,N] = A[M,K] * B[K,N] (+ optional epilogue biases)
// One wave computes one 16x16 tile using V_WMMA_F32_16X16X4_F32.
// Fragment layouts per CDNA5 ISA 7.12.2:
//   A 16x4 : lanes 0-15 -> M=lane, v0=K0,v1=K1 ; lanes 16-31 -> v0=K2,v1=K3
//   B 4x16 : VGPR r holds rows K=r (lanes 0-15) and K=r+2 (lanes 16-31)
//   C 16x16: VGPR r holds rows M=r (lanes 0-15) and M=r+8 (lanes 16-31)
// TB=true reads B as transpose of a row-major [N,K] matrix.
// ---------------------------------------------------------------------------
template <bool TB>
__global__ void wmma_gemm_f32(const float* __restrict__ A, int lda,
                              const float* __restrict__ Bm, int ldb,
                              float* __restrict__ C, int ldc,
                              int M, int N, int K,
                              const float* __restrict__ bias0,   // per-col, may be null
                              const float* __restrict__ bias1,   // per-col scaled by rowflag
                              const float* __restrict__ rowflag) // per-row
{
    const int lane = threadIdx.x & 31;
    const int wave = (blockIdx.x * blockDim.x + threadIdx.x) >> 5;
    const int ntiles = N >> 4;
    const int mt = wave / ntiles;
    const int nt = wave % ntiles;
    if (mt * 16 >= M) return;             // wave-uniform exit; EXEC stays full

    const int l16 = lane & 15;
    const int hi  = lane >> 4;            // 0 or 1
    const int arow = mt * 16 + l16;
    const int bcol = nt * 16 + l16;

    v8f acc = {};
    for (int k = 0; k < K; k += 4) {
        v2f a, b;
        a.x = A[(size_t)arow * lda + k + 2 * hi];
        a.y = A[(size_t)arow * lda + k + 2 * hi + 1];
        if (TB) {
            b.x = Bm[(size_t)bcol * ldb + k + 2 * hi];
            b.y = Bm[(size_t)bcol * ldb + k + 2 * hi + 1];
        } else {
            b.x = Bm[(size_t)(k + 2 * hi) * ldb + bcol];
            b.y = Bm[(size_t)(k + 2 * hi + 1) * ldb + bcol];
        }
        acc = __builtin_amdgcn_wmma_f32_16x16x4_f32(
            /*neg_a=*/false, a, /*neg_b=*/false, b,
            /*c_mod=*/(short)0, acc, /*reuse_a=*/false, /*reuse_b=*/false);
    }
#pragma unroll
    for (int r = 0; r < 8; ++r) {
        const int row = mt * 16 + r + 8 * hi;
        float v = acc[r];
        if (bias0) v += bias0[bcol];
        if (bias1) v += rowflag[row] * bias1[bcol];
        C[(size_t)row * ldc + bcol] = v;
    }
}

// ---------------------------------------------------------------------------
// Setup: Qs = (queries@w_q.T + b_q)*scale ;  T[h][m] = sum_d Qs[hd]*w_k[hd,m]
//        U[h][c] = sum_m T[h][m]*w_key[m,c] ; Cc[h] = T[h]·b_key + Qs_h·b_k_h
// One block, 256 threads, phased with barriers. All tiny.
// ---------------------------------------------------------------------------
__global__ void k_setup_q(const float* __restrict__ q, const float* __restrict__ w_q,
                          const float* __restrict__ b_q, const float* __restrict__ w_k,
                          const float* __restrict__ b_k, const float* __restrict__ w_key,
                          const float* __restrict__ b_key,
                          float* __restrict__ Qs, float* __restrict__ Tm,
                          float* __restrict__ U, float* __restrict__ Cc)
{
    const int t = threadIdx.x;
    float s = 0.f;
    for (int c = 0; c < EDIM; ++c) s += q[c] * w_q[t * EDIM + c];
    Qs[t] = (s + b_q[t]) * 0.125f;   // 1/sqrt(DH)=1/8
    __syncthreads();
    for (int i = t; i < HNUM * EDIM; i += 256) {
        const int h = i >> 8, m = i & 255;
        float acc = 0.f;
        for (int d = 0; d < 64; ++d) acc += Qs[h * 64 + d] * w_k[(h * 64 + d) * EDIM + m];
        Tm[i] = acc;
    }
    __syncthreads();
    for (int i = t; i < HNUM * EDIM; i += 256) {
        const int h = i >> 8, c = i & 255;
        float acc = 0.f;
        for (int m = 0; m < EDIM; ++m) acc += Tm[h * EDIM + m] * w_key[m * EDIM + c];
        U[i] = acc;
    }
    if (t < HNUM) {
        float acc = 0.f;
        for (int m = 0; m < EDIM; ++m) acc += Tm[t * EDIM + m] * b_key[m];
        for (int d = 0; d < 64; ++d) acc += Qs[t * 64 + d] * b_k[t * 64 + d];
        Cc[t] = acc;
    }
}

// bV[e] = w_v[e]·b_val + b_v[e] ;  bo_sum[e] = w_o[e]·bV
__global__ void k_setup_bias(const float* __restrict__ w_v, const float* __restrict__ b_val,
                             const float* __restrict__ b_v, const float* __restrict__ w_o,
                             float* __restrict__ bV, float* __restrict__ bo_sum)
{
    const int t = threadIdx.x;
    float s = 0.f;
    for (int m = 0; m < EDIM; ++m) s += w_v[t * EDIM + m] * b_val[m];
    bV[t] = s + b_v[t];
    __syncthreads();
    float s2 = 0.f;
    for (int m = 0; m < EDIM; ++m) s2 += w_o[t * EDIM + m] * bV[m];
    bo_sum[t] = s2;
}

// ---------------------------------------------------------------------------
// Pass 1 (bandwidth): scores[n,h] = x[n]·U[h] + Cc[h].  One wave per node,
// float4 coalesced reads, shfl_xor reduction across the wave.
// ---------------------------------------------------------------------------
__global__ void k_scores(const float* __restrict__ x, const float* __restrict__ U,
                         const float* __restrict__ Cc, float* __restrict__ scores, int Nn)
{
    __shared__ float su[HNUM * EDIM];
    __shared__ float sc[HNUM];
    const int tid = threadIdx.x;
    for (int i = tid; i < HNUM * EDIM; i += blockDim.x) su[i] = U[i];
    if (tid < HNUM) sc[tid] = Cc[tid];
    __syncthreads();

    const int lane   = tid & 31;
    const int wave   = (blockIdx.x * blockDim.x + tid) >> 5;
    const int nwaves = (gridDim.x * blockDim.x) >> 5;
    for (int n = wave; n < Nn; n += nwaves) {
        const float4* xr = (const float4*)(x + (size_t)n * EDIM);
        float a0 = 0.f, a1 = 0.f, a2 = 0.f, a3 = 0.f;
#pragma unroll
        for (int j = 0; j < 2; ++j) {
            const float4 xv = xr[lane + j * 32];
            const int c = (lane + j * 32) * 4;
            a0 += xv.x * su[0 * EDIM + c] + xv.y * su[0 * EDIM + c + 1] + xv.z * su[0 * EDIM + c + 2] + xv.w * su[0 * EDIM + c + 3];
            a1 += xv.x * su[1 * EDIM + c] + xv.y * su[1 * EDIM + c + 1] + xv.z * su[1 * EDIM + c + 2] + xv.w * su[1 * EDIM + c + 3];
            a2 += xv.x * su[2 * EDIM + c] + xv.y * su[2 * EDIM + c + 1] + xv.z * su[2 * EDIM + c + 2] + xv.w * su[2 * EDIM + c + 3];
            a3 += xv.x * su[3 * EDIM + c] + xv.y * su[3 * EDIM + c + 1] + xv.z * su[3 * EDIM + c + 2] + xv.w * su[3 * EDIM + c + 3];
        }
#pragma unroll
        for (int off = 16; off > 0; off >>= 1) {
            a0 += __shfl_xor(a0, off, 32);
            a1 += __shfl_xor(a1, off, 32);
            a2 += __shfl_xor(a2, off, 32);
            a3 += __shfl_xor(a3, off, 32);
        }
        if (lane == 0) {
            float4* so = (float4*)(scores + (size_t)n * 4);
            *so = make_float4(a0 + sc[0], a1 + sc[1], a2 + sc[2], a3 + sc[3]);
        }
    }
}

// Segment boundaries from sorted batch (no atomics, deterministic).
__global__ void k_starts(const int* __restrict__ batch, int* __restrict__ starts, int Nn, int Bg)
{
    const int n = blockIdx.x * blockDim.x + threadIdx.x;
    if (n >= Nn) return;
    const int bc = batch[n];
    const int bp = (n == 0) ? -1 : batch[n - 1];
    for (int b = bp + 1; b <= bc; ++b) starts[b] = n;
    if (n == Nn - 1)
        for (int b = bc + 1; b <= Bg; ++b) starts[b] = Nn;
}

// Per-graph softmax stats: max and 1/sum(exp) per head, plus nonempty flag.
__global__ void k_stats(const float* __restrict__ scores, const int* __restrict__ starts,
                        float* __restrict__ segmax, float* __restrict__ seginv,
                        float* __restrict__ nonempty)
{
    __shared__ float4 red[128];
    const int b = blockIdx.x, t = threadIdx.x;
    const int s = starts[b], e = starts[b + 1];

    float4 mx = make_float4(-INFINITY, -INFINITY, -INFINITY, -INFINITY);
    for (int n = s + t; n < e; n += 128) {
        const float4 v = ((const float4*)scores)[n];
        mx.x = fmaxf(mx.x, v.x); mx.y = fmaxf(mx.y, v.y);
        mx.z = fmaxf(mx.z, v.z); mx.w = fmaxf(mx.w, v.w);
    }
    red[t] = mx; __syncthreads();
    for (int off = 64; off > 0; off >>= 1) {
        if (t < off) {
            float4 o = red[t + off];
            red[t].x = fmaxf(red[t].x, o.x); red[t].y = fmaxf(red[t].y, o.y);
            red[t].z = fmaxf(red[t].z, o.z); red[t].w = fmaxf(red[t].w, o.w);
        }
        __syncthreads();
    }
    float4 M = red[0];
    M.x = isfinite(M.x) ? M.x : 0.f; M.y = isfinite(M.y) ? M.y : 0.f;
    M.z = isfinite(M.z) ? M.z : 0.f; M.w = isfinite(M.w) ? M.w : 0.f;
    __syncthreads();

    float4 sm = make_float4(0.f, 0.f, 0.f, 0.f);
    for (int n = s + t; n < e; n += 128) {
        const float4 v = ((const float4*)scores)[n];
        sm.x += expf(v.x - M.x); sm.y += expf(v.y - M.y);
        sm.z += expf(v.z - M.z); sm.w += expf(v.w - M.w);
    }
    red[t] = sm; __syncthreads();
    for (int off = 64; off > 0; off >>= 1) {
        if (t < off) {
            float4 o = red[t + off];
            red[t].x += o.x; red[t].y += o.y; red[t].z += o.z; red[t].w += o.w;
        }
        __syncthreads();
    }
    if (t == 0) {
        float4 D = red[0];
        D.x = (D.x == 0.f) ? 1.f : D.x; D.y = (D.y == 0.f) ? 1.f : D.y;
        D.z = (D.z == 0.f) ? 1.f : D.z; D.w = (D.w == 0.f) ? 1.f : D.w;
        segmax[b * 4 + 0] = M.x; segmax[b * 4 + 1] = M.y;
        segmax[b * 4 + 2] = M.z; segmax[b * 4 + 3] = M.w;
        seginv[b * 4 + 0] = 1.f / D.x; seginv[b * 4 + 1] = 1.f / D.y;
        seginv[b * 4 + 2] = 1.f / D.z; seginv[b * 4 + 3] = 1.f / D.w;
        nonempty[b] = (e > s) ? 1.f : 0.f;
    }
}

// ---------------------------------------------------------------------------
// Pass 2 (bandwidth): y[b,h,c] = sum_{n in graph b} attn[n,h] * x[n,c].
// One block per graph; thread t owns column c=t; attention weights staged in
// LDS in tiles of 16 nodes.
// ---------------------------------------------------------------------------
__global__ void k_ysum(const float* __restrict__ x, const float* __restrict__ scores,
                       const int* __restrict__ starts, const float* __restrict__ segmax,
                       const float* __restrict__ seginv, float* __restrict__ y)
{
    __shared__ float wbuf[16 * 4];
    __shared__ float smx[4], sinv[4];
    const int b = blockIdx.x, t = threadIdx.x;
    const int s = starts[b], e = starts[b + 1];
    if (t < 4) { smx[t] = segmax[b * 4 + t]; sinv[t] = seginv[b * 4 + t]; }
    __syncthreads();

    float a0 = 0.f, a1 = 0.f, a2 = 0.f, a3 = 0.f;
    for (int base = s; base < e; base += 16) {
        if (t < 64) {
            const int n = base + (t >> 2);
            const int h = t & 3;
            if (n < e) wbuf[t] = expf(scores[(size_t)n * 4 + h] - smx[h]) * sinv[h];
        }
        __syncthreads();
        const int cnt = min(16, e - base);
        for (int i = 0; i < cnt; ++i) {
            const float xv = x[(size_t)(base + i) * EDIM + t];
            a0 += wbuf[i * 4 + 0] * xv;
            a1 += wbuf[i * 4 + 1] * xv;
            a2 += wbuf[i * 4 + 2] * xv;
            a3 += wbuf[i * 4 + 3] * xv;
        }
        __syncthreads();
    }
    y[(size_t)(b * 4 + 0) * EDIM + t] = a0;
    y[(size_t)(b * 4 + 1) * EDIM + t] = a1;
    y[(size_t)(b * 4 + 2) * EDIM + t] = a2;
    y[(size_t)(b * 4 + 3) * EDIM + t] = a3;
}

// ---------------------------------------------------------------------------
extern "C" void kernel_launch(void* const* d_in, const int* in_sizes, int n_in,
                              void* d_out, int out_size, void* d_ws, size_t ws_size,
                              hipStream_t stream)
{
    const float* x     = (const float*)d_in[0];
    const float* q     = (const float*)d_in[1];
    const float* w_key = (const float*)d_in[2];
    const float* b_key = (const float*)d_in[3];
    const float* w_val = (const float*)d_in[4];
    const float* b_val = (const float*)d_in[5];
    const float* w_q   = (const float*)d_in[6];
    const float* b_q   = (const float*)d_in[7];
    const float* w_k   = (const float*)d_in[8];
    const float* b_k   = (const float*)d_in[9];
    const float* w_v   = (const float*)d_in[10];
    const float* b_v   = (const float*)d_in[11];
    const float* w_o   = (const float*)d_in[12];
    const float* b_o   = (const float*)d_in[13];
    const int*   batch = (const int*)d_in[14];
    const int Nn = in_sizes[14];            // number of nodes
    const int Bg = BGRAPH;                  // fixed by reference

    // Workspace layout (floats)
    float* w = (float*)d_ws;
    float* Qs    = w;                 // 256
    float* Tm    = Qs + 256;          // 1024
    float* U     = Tm + 1024;         // 1024
    float* Cc    = U + 1024;          // 4 (pad 16)
    float* Wv    = Cc + 16;           // 256*256
    float* bV    = Wv + 65536;        // 256
    float* bosum = bV + 256;          // 256
    float* Wf    = bosum + 256;       // 256*1024
    float* smax  = Wf + 262144;       // B*4
    float* sinv  = smax + BGRAPH * 4; // B*4
    float* nemp  = sinv + BGRAPH * 4; // B
    float* y     = nemp + BGRAPH;     // B*4*256
    float* scr   = y + BGRAPH * KCOMB;// N*4
    int*   starts = (int*)(scr + (size_t)Nn * 4); // B+1

    // --- tiny setup ---
    k_setup_q<<<1, 256, 0, stream>>>(q, w_q, b_q, w_k, b_k, w_key, b_key, Qs, Tm, U, Cc);
    k_setup_bias<<<1, 256, 0, stream>>>(w_v, b_val, b_v, w_o, bV, bosum);

    // --- Wv = w_v @ w_val  (256x256x256, f32 WMMA) ---
    // tiles = 16*16 = 256 waves -> 32 blocks of 8 waves
    wmma_gemm_f32<false><<<32, 256, 0, stream>>>(w_v, EDIM, w_val, EDIM, Wv, EDIM,
                                                 EDIM, EDIM, EDIM, nullptr, nullptr, nullptr);

    // --- Wf[e, h*256+c] = sum_d w_o[e,h*64+d] * Wv[h*64+d, c]  (4x 256x256x64) ---
    for (int h = 0; h < HNUM; ++h) {
        wmma_gemm_f32<false><<<32, 256, 0, stream>>>(w_o + h * 64, EDIM,
                                                     Wv + (size_t)h * 64 * EDIM, EDIM,
                                                     Wf + h * EDIM, KCOMB,
                                                     EDIM, EDIM, 64, nullptr, nullptr, nullptr);
    }

    // --- pass 1: scores ---
    k_scores<<<1024, 256, 0, stream>>>(x, U, Cc, scr, Nn);

    // --- segment boundaries + softmax stats ---
    k_starts<<<(Nn + 255) / 256, 256, 0, stream>>>(batch, starts, Nn, Bg);
    k_stats<<<Bg, 128, 0, stream>>>(scr, starts, smax, sinv, nemp);

    // --- pass 2: weighted segment sums y ---
    k_ysum<<<Bg, 256, 0, stream>>>(x, scr, starts, smax, sinv, y);

    // --- out = y @ Wf.T + b_o + nonempty*bo_sum  (512x256x1024, f32 WMMA) ---
    // tiles = 32*16 = 512 waves -> 64 blocks
    wmma_gemm_f32<true><<<64, 256, 0, stream>>>(y, KCOMB, Wf, KCOMB, (float*)d_out, EDIM,
                                                BGRAPH, EDIM, KCOMB, b_o, bosum, nemp);
}